// STFFilterLayer_80891414053019
// MI455X (gfx1250) — compile-verified
//
#include <hip/hip_runtime.h>
#include <hip/hip_bf16.h>

// ============================================================================
// STFT filter layer on MI455X (gfx1250).
// All three linear stages expressed as fp32 WMMA GEMMs (V_WMMA_F32_16X16X4_F32,
// wave32).  Stage 2 (per-frequency complex channel mixing, the dominant GEMM)
// stages its operands in LDS via the Tensor Data Mover (tensor_load_to_lds +
// s_wait_tensorcnt) and feeds WMMA from ds_load.
//
//   stage 1: XR/XI[bf, fpad, 128]  = DFTcos/DFTsin[fpad, n] @ x_frame[n, 128]
//   stage 2: per bin f: YR/YI[bf,64] = complex( X[bf,128] @ K[f,128,64] ), lrelu
//   stage 3: out slice = ISYNr[n, fpad] @ YR + ISYNi[n, fpad] @ YI  (+ bias)
//
// fpad = round16(n/2+1) with zeroed table rows beyond n/2, so every GEMM dim
// is a multiple of the 16x16x4 tile and EXEC stays all-ones around WMMA.
// ============================================================================

typedef float v2f __attribute__((ext_vector_type(2)));
typedef float v8f __attribute__((ext_vector_type(8)));
typedef unsigned int u32x4 __attribute__((ext_vector_type(4)));
typedef int i32x4 __attribute__((ext_vector_type(4)));
typedef int i32x8 __attribute__((ext_vector_type(8)));

#define LRELU_ALPHA 0.2f
#define CI 128           // input channels
#define CE 64            // channels per resolution
#define MTILE 32         // stage-2 M rows per workgroup (2 waves x 16)

// ---- WMMA f32 16x16x4 fragment helpers (wave32 layouts per CDNA5 ISA) -----
// A 16x4 : lanes 0-15 hold rows M=0..15, K={0,1}; lanes 16-31 hold K={2,3}
// B 4x16 : lanes 0-15 hold cols N=0..15, K={0,1}; lanes 16-31 hold K={2,3}
// C/D    : VGPR j -> M=j (lanes 0-15) / M=j+8 (lanes 16-31), N = lane&15

__device__ __forceinline__ v2f frag_a(const float* __restrict__ p, int rowStride, int lane) {
  const int m  = lane & 15;
  const int kb = (lane >> 4) << 1;
  v2f a;
  a.x = p[(size_t)m * rowStride + kb];
  a.y = p[(size_t)m * rowStride + kb + 1];
  return a;
}

__device__ __forceinline__ v2f frag_b(const float* __restrict__ p, int rowStride, int lane) {
  const int nn = lane & 15;
  const int kb = (lane >> 4) << 1;
  v2f b;
  b.x = p[(size_t)kb * rowStride + nn];
  b.y = p[(size_t)(kb + 1) * rowStride + nn];
  return b;
}

#define WMMA_F32(A, B, C) \
  __builtin_amdgcn_wmma_f32_16x16x4_f32(false, (A), false, (B), (short)0, (C), false, false)

// ---------------------------------------------------------------------------
// Tensor Data Mover: issue one D# describing a [rows x rowlen] fp32 tile
// (row-major, row stride in elements) copied global -> LDS.  Descriptor bit
// packing follows CDNA5 ISA 8.3/8.4 (group0: count|lds_addr|global_addr|type;
// group1: data_size, tensor dims, tile dims, 48-bit dim0 stride).  Groups 2/3
// are unused for <=2D tiles.
// ---------------------------------------------------------------------------
__device__ __forceinline__ void tdm_load_tile(const void* gptr, unsigned lds_off,
                                              unsigned rowlen, unsigned rows,
                                              unsigned long long row_stride_elems) {
  const unsigned long long ga = (unsigned long long)(size_t)gptr;

  u32x4 g0;
  g0.x = 1u;                                                  // count=1, user mode
  g0.y = lds_off;                                             // lds_addr (bytes)
  g0.z = (unsigned)(ga & 0xffffffffull);                      // global_addr[31:0]
  g0.w = (unsigned)((ga >> 32) & 0x01ffffffull) | (2u << 30); // addr[56:32] | type=2

  i32x8 g1;
  g1[0] = (int)(2u << 16);                                    // data_size = 4 bytes
  g1[1] = (int)((rowlen & 0xffffu) << 16);                    // tensor_dim0 lo16
  g1[2] = (int)(((rowlen >> 16) & 0xffffu) |                  // tensor_dim0 hi16
                ((rows & 0xffffu) << 16));                    // tensor_dim1 lo16
  g1[3] = (int)(((rows >> 16) & 0xffffu) |                    // tensor_dim1 hi16
                ((rowlen & 0xffffu) << 16));                  // tile_dim0
  g1[4] = (int)(rows & 0xffffu);                              // tile_dim1 (tile_dim2=0)
  g1[5] = (int)(row_stride_elems & 0xffffffffull);            // dim0_stride lo32
  g1[6] = (int)((row_stride_elems >> 32) & 0xffffull);        // dim0_stride hi16
  g1[7] = 0;

  i32x4 z4 = {};
#if defined(__clang_major__) && (__clang_major__ >= 23)
  i32x8 z8 = {};
  __builtin_amdgcn_tensor_load_to_lds(g0, g1, z4, z4, z8, 0);
#else
  __builtin_amdgcn_tensor_load_to_lds(g0, g1, z4, z4, 0);
#endif
}

#define LDS_OFF(p) ((unsigned)(size_t)(p))  // flat shared addr: low 32 bits = LDS offset

// ---------------------------------------------------------------------------
// DFT / inverse-synthesis table builder.
//   fwdC[f*n + t] =  cos(2*pi*f*t/n)            (f <= n/2, else 0)
//   fwdS[f*n + t] = -sin(2*pi*f*t/n)            (imag accumulates -sin)
//   invR[t*fpad+f] =  (w_f/n)*cos(2*pi*f*t/n)   w_f = 1 for f in {0,n/2}, else 2
//   invI[t*fpad+f] = -(w_f/n)*sin(2*pi*f*t/n)
// ---------------------------------------------------------------------------
__global__ void stf_dft_tables_kernel(float* __restrict__ fwdC, float* __restrict__ fwdS,
                                      float* __restrict__ invR, float* __restrict__ invI,
                                      int n, int fpad) {
  int idx = blockIdx.x * blockDim.x + threadIdx.x;
  int total = fpad * n;
  if (idx >= total) return;
  int f = idx / n;
  int t = idx - f * n;
  int half = n >> 1;
  float c = 0.0f, s = 0.0f;
  if (f <= half) {
    int r = (f * t) % n;  // keep the angle small for fp32 accuracy
    float theta = 6.2831853071795864769f * (float)r / (float)n;
    __sincosf(theta, &s, &c);
  }
  fwdC[(size_t)f * n + t] = c;
  fwdS[(size_t)f * n + t] = -s;
  float w = (f == 0 || f == half) ? 1.0f : 2.0f;
  float scale = w / (float)n;
  invR[(size_t)t * fpad + f] = c * scale;
  invI[(size_t)t * fpad + f] = -s * scale;
}

// ---------------------------------------------------------------------------
// Stage 1: forward DFT as GEMM.  One wave per 16(f) x 64(ci) tile of a frame;
// the cos/sin A-fragments are reused across the four 16-wide N tiles.
// x viewed as [B*frames][n][ci] (frames are contiguous along T).
// ---------------------------------------------------------------------------
__global__ __launch_bounds__(32)
void stf_fwd_dft_kernel(const float* __restrict__ x,
                        const float* __restrict__ fwdC, const float* __restrict__ fwdS,
                        float* __restrict__ XR, float* __restrict__ XI,
                        int n, int fpad, int ci) {
  const int lane  = threadIdx.x;
  const int frow0 = blockIdx.x * 16;   // frequency-row tile
  const int cib   = blockIdx.y * 64;   // channel tile (64 wide)
  const int bf    = blockIdx.z;        // frame index (b*frames + fr)

  const float* aC = fwdC + (size_t)frow0 * n;
  const float* aS = fwdS + (size_t)frow0 * n;
  const float* bX = x + (size_t)bf * n * ci + cib;

  v8f accR[4] = {};
  v8f accI[4] = {};
  for (int k = 0; k < n; k += 4) {
    v2f ac = frag_a(aC + k, n, lane);
    v2f as = frag_a(aS + k, n, lane);
    #pragma unroll
    for (int j = 0; j < 4; ++j) {
      v2f bx = frag_b(bX + (size_t)k * ci + j * 16, ci, lane);
      accR[j] = WMMA_F32(ac, bx, accR[j]);
      accI[j] = WMMA_F32(as, bx, accI[j]);
    }
  }

  const int M = (lane >> 4) << 3;
  const int N = lane & 15;
  #pragma unroll
  for (int j = 0; j < 4; ++j) {
    size_t obase = ((size_t)bf * fpad + frow0) * ci + cib + j * 16 + N;
    #pragma unroll
    for (int jj = 0; jj < 8; ++jj) {
      XR[obase + (size_t)(jj + M) * ci] = accR[j][jj];
      XI[obase + (size_t)(jj + M) * ci] = accI[j][jj];
    }
  }
}

// ---------------------------------------------------------------------------
// Stage 2: per-frequency complex channel mixing + leaky ReLU.
//   YR = XR@Kr - XI@Ki ;  YI = XR@Ki + XI@Kr   (M = B*frames, K = 128, N = 64)
// TDM stages XR/XI (2-D row-strided tiles) and Kr/Ki (contiguous bin slices)
// into LDS; 2 waves each compute a 16x64 slice with 16 WMMAs per K-step.
// Padded bins f >= fbins zero the K tiles (uniform branch) -> Y = 0.
// ---------------------------------------------------------------------------
__global__ __launch_bounds__(64)
void stf_mix_kernel(const float* __restrict__ XR, const float* __restrict__ XI,
                    const float* __restrict__ Kr, const float* __restrict__ Ki,
                    float* __restrict__ YR, float* __restrict__ YI,
                    int fpad, int fbins) {
  __shared__ float sXR[MTILE * CI];   // 16 KB
  __shared__ float sXI[MTILE * CI];   // 16 KB
  __shared__ float sKR[CI * CE];      // 32 KB
  __shared__ float sKI[CI * CE];      // 32 KB

  const int tid  = threadIdx.x;
  const int lane = tid & 31;
  const int wv   = tid >> 5;          // wave id: 0..1
  const int f    = blockIdx.x;        // frequency bin (uniform per block)
  const int m0   = blockIdx.y * MTILE;
  const bool valid = (f < fbins);

  if (wv == 0) {
    const unsigned long long rs = (unsigned long long)fpad * (unsigned long long)CI;
    tdm_load_tile(XR + ((size_t)m0 * fpad + f) * CI, LDS_OFF(sXR), CI, MTILE, rs);
    tdm_load_tile(XI + ((size_t)m0 * fpad + f) * CI, LDS_OFF(sXI), CI, MTILE, rs);
    if (valid) {
      tdm_load_tile(Kr + (size_t)f * CI * CE, LDS_OFF(sKR), CI * CE, 1, CI * CE);
      tdm_load_tile(Ki + (size_t)f * CI * CE, LDS_OFF(sKI), CI * CE, 1, CI * CE);
    }
    __builtin_amdgcn_s_wait_tensorcnt(0);
  }
  if (!valid) {
    for (int i = tid; i < CI * CE; i += 64) { sKR[i] = 0.0f; sKI[i] = 0.0f; }
  }
  __syncthreads();

  const int mrow = wv * 16;           // this wave's M offset within the tile
  const int am   = mrow + (lane & 15);
  const int kb   = (lane >> 4) << 1;
  const int nn   = lane & 15;

  v8f aRR[4] = {};   // XR@Kr
  v8f aII[4] = {};   // XI@Ki
  v8f aMX[4] = {};   // XR@Ki + XI@Kr
  for (int k = 0; k < CI; k += 4) {
    v2f ar, ai;
    ar.x = sXR[am * CI + k + kb];
    ar.y = sXR[am * CI + k + kb + 1];
    ai.x = sXI[am * CI + k + kb];
    ai.y = sXI[am * CI + k + kb + 1];
    #pragma unroll
    for (int j = 0; j < 4; ++j) {
      v2f br, bi;
      br.x = sKR[(k + kb) * CE + j * 16 + nn];
      br.y = sKR[(k + kb + 1) * CE + j * 16 + nn];
      bi.x = sKI[(k + kb) * CE + j * 16 + nn];
      bi.y = sKI[(k + kb + 1) * CE + j * 16 + nn];
      aRR[j] = WMMA_F32(ar, br, aRR[j]);
      aII[j] = WMMA_F32(ai, bi, aII[j]);
      aMX[j] = WMMA_F32(ar, bi, aMX[j]);
      aMX[j] = WMMA_F32(ai, br, aMX[j]);
    }
  }

  const int M = (lane >> 4) << 3;
  #pragma unroll
  for (int j = 0; j < 4; ++j) {
    #pragma unroll
    for (int jj = 0; jj < 8; ++jj) {
      float yr = aRR[j][jj] - aII[j][jj];
      float yi = aMX[j][jj];
      yr = (yr >= 0.0f) ? yr : LRELU_ALPHA * yr;   // freq-domain leaky ReLU
      yi = (yi >= 0.0f) ? yi : LRELU_ALPHA * yi;
      size_t oidx = (((size_t)(m0 + mrow + jj + M)) * fpad + f) * CE + j * 16 + nn;
      YR[oidx] = yr;
      YI[oidx] = yi;
    }
  }
}

// ---------------------------------------------------------------------------
// Stage 3: inverse DFT as GEMM + bias + channel-concat store.
// One wave per 16(t) x 64(ce) tile; synthesis A-fragments reused across N.
// ---------------------------------------------------------------------------
__global__ __launch_bounds__(32)
void stf_inv_dft_kernel(const float* __restrict__ YR, const float* __restrict__ YI,
                        const float* __restrict__ invR, const float* __restrict__ invI,
                        const float* __restrict__ bias, float* __restrict__ out,
                        int n, int fpad, int co, int cout) {
  const int lane = threadIdx.x;
  const int t0   = blockIdx.x * 16;   // time tile within frame
  const int bf   = blockIdx.y;        // frame index

  const float* ar = invR + (size_t)t0 * fpad;
  const float* ai = invI + (size_t)t0 * fpad;
  const float* yr = YR + (size_t)bf * fpad * CE;
  const float* yi = YI + (size_t)bf * fpad * CE;

  v8f acc[4] = {};
  for (int k = 0; k < fpad; k += 4) {
    v2f a1 = frag_a(ar + k, fpad, lane);
    v2f a2 = frag_a(ai + k, fpad, lane);
    if (k + 8 < fpad) {  // pull next K rows toward the caches (global_prefetch_b8)
      __builtin_prefetch(yr + (size_t)(k + 8) * CE, 0, 1);
      __builtin_prefetch(yi + (size_t)(k + 8) * CE, 0, 1);
    }
    #pragma unroll
    for (int j = 0; j < 4; ++j) {
      v2f b1 = frag_b(yr + (size_t)k * CE + j * 16, CE, lane);
      v2f b2 = frag_b(yi + (size_t)k * CE + j * 16, CE, lane);
      acc[j] = WMMA_F32(a1, b1, acc[j]);
      acc[j] = WMMA_F32(a2, b2, acc[j]);
    }
  }

  const int M = (lane >> 4) << 3;
  const int N = lane & 15;
  #pragma unroll
  for (int j = 0; j < 4; ++j) {
    const float bv = bias[co + j * 16 + N];
    // global time index of row t within frame bf is bf*n + t; out is [B*T, cout]
    size_t obase = ((size_t)bf * n + t0) * cout + co + j * 16 + N;
    #pragma unroll
    for (int jj = 0; jj < 8; ++jj) {
      out[obase + (size_t)(jj + M) * cout] = acc[j][jj] + bv;
    }
  }
}

// ---------------------------------------------------------------------------
// Launcher: three resolutions processed sequentially on `stream`, reusing one
// workspace region (tables + XR/XI + YR/YI).
// ---------------------------------------------------------------------------
extern "C" void kernel_launch(void* const* d_in, const int* in_sizes, int n_in,
                              void* d_out, int out_size, void* d_ws, size_t ws_size,
                              hipStream_t stream) {
  (void)in_sizes; (void)n_in; (void)out_size; (void)ws_size;

  const float* x    = (const float*)d_in[0];
  const float* bias = (const float*)d_in[7];
  float* out = (float*)d_out;

  const int B = 32, T = 8192, ci = CI, cout = 192, ce = CE;

  for (int r = 0; r < 3; ++r) {
    const int n      = 128 << r;           // 128, 256, 512
    const int frames = T / n;
    const int BF     = B * frames;         // 2048, 1024, 512
    const int fbins  = n / 2 + 1;          // 65, 129, 257
    const int fpad   = (fbins + 15) & ~15; // 80, 144, 272
    const int co     = r * ce;             // channel offset in concatenated out

    const float* Kr = (const float*)d_in[1 + 2 * r];
    const float* Ki = (const float*)d_in[2 + 2 * r];

    // Workspace carve-out (floats), reused across resolutions (stream-ordered).
    float* w = (float*)d_ws;
    const size_t tab = (size_t)fpad * n;
    float* fwdC = w;              w += tab;
    float* fwdS = w;              w += tab;
    float* invRt = w;             w += tab;
    float* invIt = w;             w += tab;
    const size_t xsz = (size_t)BF * fpad * ci;
    float* XR = w;                w += xsz;
    float* XI = w;                w += xsz;
    const size_t ysz = (size_t)BF * fpad * ce;
    float* YR = w;                w += ysz;
    float* YI = w;                w += ysz;

    // 0) basis tables
    {
      int total = fpad * n;
      int blocks = (total + 255) / 256;
      stf_dft_tables_kernel<<<blocks, 256, 0, stream>>>(fwdC, fwdS, invRt, invIt, n, fpad);
    }
    // 1) forward DFT (rfft as GEMM)
    stf_fwd_dft_kernel<<<dim3(fpad / 16, ci / 64, BF), 32, 0, stream>>>(
        x, fwdC, fwdS, XR, XI, n, fpad, ci);
    // 2) per-bin complex channel mixing + leaky ReLU (TDM -> LDS -> WMMA)
    stf_mix_kernel<<<dim3(fpad, BF / MTILE), 64, 0, stream>>>(
        XR, XI, Kr, Ki, YR, YI, fpad, fbins);
    // 3) inverse DFT + bias + concat store
    stf_inv_dft_kernel<<<dim3(n / 16, BF), 32, 0, stream>>>(
        YR, YI, invRt, invIt, bias, out, n, fpad, co, cout);
  }
}